// RotationInvarianceTorch_84997402788331
// MI455X (gfx1250) — compile-verified
//
#include <hip/hip_runtime.h>

typedef __attribute__((ext_vector_type(2))) float v2f;
typedef __attribute__((ext_vector_type(8))) float v8f;

#define NPTS  2048
#define TPB   256
#define WAVES (TPB / 32)

// Null-space eigenvector of symmetric [a d f; d b e; f e c] for eigenvalue lam,
// via the largest cross product of rows of (A - lam I).
__device__ __forceinline__ void nullspace_vec(float a, float b, float c,
                                              float d, float e, float f,
                                              float lam, float v[3]) {
  float r0x = a - lam, r0y = d,       r0z = f;
  float r1x = d,       r1y = b - lam, r1z = e;
  float r2x = f,       r2y = e,       r2z = c - lam;
  float c0x = r0y * r1z - r0z * r1y, c0y = r0z * r1x - r0x * r1z, c0z = r0x * r1y - r0y * r1x;
  float c1x = r0y * r2z - r0z * r2y, c1y = r0z * r2x - r0x * r2z, c1z = r0x * r2y - r0y * r2x;
  float c2x = r1y * r2z - r1z * r2y, c2y = r1z * r2x - r1x * r2z, c2z = r1x * r2y - r1y * r2x;
  float n0 = c0x * c0x + c0y * c0y + c0z * c0z;
  float n1 = c1x * c1x + c1y * c1y + c1z * c1z;
  float n2 = c2x * c2x + c2y * c2y + c2z * c2z;
  float vx = c0x, vy = c0y, vz = c0z, nb = n0;
  if (n1 > nb) { vx = c1x; vy = c1y; vz = c1z; nb = n1; }
  if (n2 > nb) { vx = c2x; vy = c2y; vz = c2z; nb = n2; }
  float inv = rsqrtf(nb + 1e-30f);
  v[0] = vx * inv; v[1] = vy * inv; v[2] = vz * inv;
}

__global__ void __launch_bounds__(TPB) pca_align_kernel(
    const float* __restrict__ pts,
    float* __restrict__ out_aligned,
    float* __restrict__ out_evals) {
  __shared__ float4 s_pts4[NPTS * 3 / 4];   // 24 KB point tile
  __shared__ float  s_cov[8];               // 6 covariance accumulators
  __shared__ float  s_stage[WAVES][96];     // per-wave D-matrix repack buffer

  const int tid = threadIdx.x;
  const int bid = blockIdx.x;
  float* sp = (float*)s_pts4;

  if (tid < 8) s_cov[tid] = 0.0f;

  // -------- Phase 1: coalesced global -> LDS copy (read HBM exactly once) ----
  const float4* g4 = (const float4*)(pts + (size_t)bid * (NPTS * 3));
  #pragma unroll
  for (int i = 0; i < (NPTS * 3 / 4) / TPB; ++i)
    s_pts4[tid + i * TPB] = g4[tid + i * TPB];

  __syncthreads();

  // -------- Phase 2: covariance reduction ----------------------------------
  float xx = 0.f, xy = 0.f, xz = 0.f, yy = 0.f, yz = 0.f, zz = 0.f;
  #pragma unroll
  for (int i = 0; i < NPTS / TPB; ++i) {
    int idx = tid + i * TPB;
    float x = sp[idx * 3 + 0], y = sp[idx * 3 + 1], z = sp[idx * 3 + 2];
    xx += x * x; xy += x * y; xz += x * z;
    yy += y * y; yz += y * z; zz += z * z;
  }
  #pragma unroll
  for (int off = 16; off > 0; off >>= 1) {
    xx += __shfl_xor(xx, off, 32);  xy += __shfl_xor(xy, off, 32);
    xz += __shfl_xor(xz, off, 32);  yy += __shfl_xor(yy, off, 32);
    yz += __shfl_xor(yz, off, 32);  zz += __shfl_xor(zz, off, 32);
  }
  if ((tid & 31) == 0) {
    atomicAdd(&s_cov[0], xx); atomicAdd(&s_cov[1], xy); atomicAdd(&s_cov[2], xz);
    atomicAdd(&s_cov[3], yy); atomicAdd(&s_cov[4], yz); atomicAdd(&s_cov[5], zz);
  }
  __syncthreads();

  // -------- Phase 3: analytic 3x3 symmetric eigendecomposition (redundant) --
  const float invN = 1.0f / (float)NPTS;
  float a  = s_cov[0] * invN + 1e-8f;
  float bb = s_cov[3] * invN + 1e-8f;
  float cc = s_cov[5] * invN + 1e-8f;
  float dd = s_cov[1] * invN;   // xy
  float ee = s_cov[4] * invN;   // yz
  float ff = s_cov[2] * invN;   // xz

  float q  = (a + bb + cc) * (1.0f / 3.0f);
  float p1 = dd * dd + ff * ff + ee * ee;
  float aq = a - q, bq = bb - q, cq = cc - q;
  float p2 = aq * aq + bq * bq + cq * cq + 2.0f * p1;
  float p  = sqrtf(p2 * (1.0f / 6.0f));

  float l1, l2, l3;
  float E[3][3];  // columns = eigenvectors, descending eigenvalue order
  if (p < 1e-20f) {
    l1 = l2 = l3 = q;
    E[0][0] = 1.f; E[0][1] = 0.f; E[0][2] = 0.f;
    E[1][0] = 0.f; E[1][1] = 1.f; E[1][2] = 0.f;
    E[2][0] = 0.f; E[2][1] = 0.f; E[2][2] = 1.f;
  } else {
    float ip  = 1.0f / p;
    float b00 = aq * ip, b11 = bq * ip, b22 = cq * ip;
    float b01 = dd * ip, b12 = ee * ip, b02 = ff * ip;
    float detB = b00 * (b11 * b22 - b12 * b12)
               - b01 * (b01 * b22 - b12 * b02)
               + b02 * (b01 * b12 - b11 * b02);
    float r = 0.5f * detB;
    r = fminf(1.0f, fmaxf(-1.0f, r));
    float phi = acosf(r) * (1.0f / 3.0f);
    l1 = q + 2.0f * p * cosf(phi);
    l3 = q + 2.0f * p * cosf(phi + 2.0943951023931953f);  // +2pi/3
    l2 = 3.0f * q - l1 - l3;

    float v1[3], v3[3];
    nullspace_vec(a, bb, cc, dd, ee, ff, l1, v1);
    nullspace_vec(a, bb, cc, dd, ee, ff, l3, v3);
    // middle vector = v3 x v1  =>  det(E) = +1 (matches reference's det>0 fix)
    float v2x = v3[1] * v1[2] - v3[2] * v1[1];
    float v2y = v3[2] * v1[0] - v3[0] * v1[2];
    float v2z = v3[0] * v1[1] - v3[1] * v1[0];
    float inv2 = rsqrtf(v2x * v2x + v2y * v2y + v2z * v2z + 1e-30f);
    E[0][0] = v1[0];      E[1][0] = v1[1];      E[2][0] = v1[2];
    E[0][1] = v2x * inv2; E[1][1] = v2y * inv2; E[2][1] = v2z * inv2;
    E[0][2] = v3[0];      E[1][2] = v3[1];      E[2][2] = v3[2];
  }

  if (tid == 0) {
    out_evals[(size_t)bid * 3 + 0] = l1;
    out_evals[(size_t)bid * 3 + 1] = l2;
    out_evals[(size_t)bid * 3 + 2] = l3;
  }

  // -------- Phase 4: aligned = P x E ---------------------------------------
  const int  lane = tid & 31;
  const int  w    = tid >> 5;
  const int  l16  = lane & 15;
  const bool lo   = lane < 16;
  float* stage = &s_stage[w][0];
  float* ob    = out_aligned + (size_t)bid * (NPTS * 3);

#if __has_builtin(__builtin_amdgcn_wmma_f32_16x16x4_f32)
  // B operand (4x16): lanes 0-15 carry rows K=0 (VGPR0) / K=1 (VGPR1),
  // lanes 16-31 carry rows K=2 (VGPR0) / K=3=zero-pad (VGPR1).
  v2f bop; bop.x = 0.f; bop.y = 0.f;
  if (l16 < 3) {
    if (lo) { bop.x = E[0][l16]; bop.y = E[1][l16]; }
    else    { bop.x = E[2][l16]; }
  }

  for (int it = 0; it < NPTS / (WAVES * 32); ++it) {
    const int m0 = w * (NPTS / WAVES) + it * 32;
    #pragma unroll
    for (int t = 0; t < 2; ++t) {
      const int base = m0 + t * 16;
      const int r    = base + l16;
      // A operand (16x4): lanes 0-15 = (x,y) at K=0/1; lanes 16-31 = (z,0) at K=2/3.
      v2f aop;
      if (lo) { aop.x = sp[r * 3 + 0]; aop.y = sp[r * 3 + 1]; }
      else    { aop.x = sp[r * 3 + 2]; aop.y = 0.f; }
      v8f dm = {};
      dm = __builtin_amdgcn_wmma_f32_16x16x4_f32(false, aop, false, bop,
                                                 (short)0, dm, false, false);
      // D layout: VGPR i, lanes 0-15 -> (M=i, N=lane); lanes 16-31 -> (M=8+i, N=lane-16).
      if (l16 < 3) {
        const int rb = t * 16 + (lo ? 0 : 8);
        #pragma unroll
        for (int i = 0; i < 8; ++i)
          stage[(rb + i) * 3 + l16] = dm[i];
      }
    }
    __syncthreads();
    // Coalesced store: lane emits 3 consecutive floats -> 384 B contiguous per wave.
    float o0 = stage[lane * 3 + 0];
    float o1 = stage[lane * 3 + 1];
    float o2 = stage[lane * 3 + 2];
    const size_t off = (size_t)(m0 + lane) * 3;
    ob[off + 0] = o0; ob[off + 1] = o1; ob[off + 2] = o2;
    __syncthreads();
  }
#else
  // Scalar fallback (should not trigger on gfx1250 toolchains with this builtin).
  (void)stage;
  for (int it = 0; it < NPTS / TPB; ++it) {
    const int m = tid + it * TPB;
    float x = sp[m * 3 + 0], y = sp[m * 3 + 1], z = sp[m * 3 + 2];
    const size_t off = (size_t)m * 3;
    ob[off + 0] = x * E[0][0] + y * E[1][0] + z * E[2][0];
    ob[off + 1] = x * E[0][1] + y * E[1][1] + z * E[2][1];
    ob[off + 2] = x * E[0][2] + y * E[1][2] + z * E[2][2];
  }
#endif
}

extern "C" void kernel_launch(void* const* d_in, const int* in_sizes, int n_in,
                              void* d_out, int out_size, void* d_ws, size_t ws_size,
                              hipStream_t stream) {
  (void)n_in; (void)out_size; (void)d_ws; (void)ws_size;
  const float* pts = (const float*)d_in[0];
  float* out = (float*)d_out;
  const int Bn = in_sizes[0] / (NPTS * 3);   // 8192 batches
  float* out_aligned = out;
  float* out_evals   = out + (size_t)Bn * NPTS * 3;
  if (Bn > 0) {
    pca_align_kernel<<<Bn, TPB, 0, stream>>>(pts, out_aligned, out_evals);
  }
}